// YoloV2LossV2_26749056320393
// MI455X (gfx1250) — compile-verified
//
#include <hip/hip_runtime.h>
#include <math.h>

// ---------------- problem constants ----------------
#define NCLS   80
#define NA     5
#define HG     19
#define WG     19
#define BATCH  128
#define TBOX   50
#define PLANE  (HG*WG)                 // 361
#define NCELL  (BATCH*NA*PLANE)        // 231040
#define CBLK   256
#define NBLK   ((NCELL + CBLK - 1)/CBLK)   // 903
#define EPSF   1e-7f
#define FOUR_OVER_PI2 0.40528473456935109f

typedef __attribute__((ext_vector_type(2))) float v2f;
typedef __attribute__((ext_vector_type(8))) float v8f;

// ---------------- CDNA5 WMMA wave reduction ----------------
// D = A(16x4) * B(4x16) + C.  B = all-ones => D[m][n] = sum_k A[m][k].
// Each lane owns exactly 2 of the 64 A slots (2 VGPRs); put v in slot0, 0 in
// slot1 => sum over all A slots == sum over 32 lanes, independent of the exact
// (bijective) lane<->(m,k) mapping.  Per the documented f32 C/D layout, lane n
// holds rows {0..7} (n<16) or {8..15} (n>=16) of column n%16, so
// sum(D vgprs) + shfl_xor(16) == full 16-row column sum == wave total (exact:
// fp32 multiplies by 1.0, fp32 adds).
__device__ __forceinline__ float wave_sum32(float v) {
  v2f a = {v, 0.0f};
  v2f b = {1.0f, 1.0f};
  v8f c = {};
  c = __builtin_amdgcn_wmma_f32_16x16x4_f32(false, a, false, b,
                                            (short)0, c, false, false);
  float s = ((c[0] + c[1]) + (c[2] + c[3])) + ((c[4] + c[5]) + (c[6] + c[7]));
  s += __shfl_xor(s, 16, 32);
  return s;
}

// ---------------- math helpers ----------------
__device__ __forceinline__ float sigm(float x) {
  return 1.0f / (1.0f + expf(-x));
}
__device__ __forceinline__ float bce(float p, float t) {
  p = fminf(fmaxf(p, 1e-12f), 1.0f - 1e-12f);
  return -(t * logf(p) + (1.0f - t) * log1pf(-p));
}
__device__ float ciou_fn(float x1, float y1, float w1, float h1,
                         float x2, float y2, float w2, float h2) {
  float b1x1 = x1 - w1 * 0.5f, b1x2 = x1 + w1 * 0.5f;
  float b1y1 = y1 - h1 * 0.5f, b1y2 = y1 + h1 * 0.5f;
  float b2x1 = x2 - w2 * 0.5f, b2x2 = x2 + w2 * 0.5f;
  float b2y1 = y2 - h2 * 0.5f, b2y2 = y2 + h2 * 0.5f;
  float iw = fmaxf(fminf(b1x2, b2x2) - fmaxf(b1x1, b2x1), 0.0f);
  float ih = fmaxf(fminf(b1y2, b2y2) - fmaxf(b1y1, b2y1), 0.0f);
  float inter = iw * ih;
  float uni = w1 * h1 + w2 * h2 - inter + EPSF;
  float iou = inter / uni;
  float cw = fmaxf(b1x2, b2x2) - fminf(b1x1, b2x1);
  float ch = fmaxf(b1y2, b2y2) - fminf(b1y1, b2y1);
  float c2 = cw * cw + ch * ch + EPSF;
  float rho2 = (x2 - x1) * (x2 - x1) + (y2 - y1) * (y2 - y1);
  float dv = atanf(w2 / (h2 + EPSF)) - atanf(w1 / (h1 + EPSF));
  float v = FOUR_OVER_PI2 * dv * dv;
  float alpha = v / (v - iou + 1.0f + EPSF);
  return iou - (rho2 / c2 + v * alpha);
}

// ---------------- kernel A: init workspace ----------------
__global__ void yolo_init_kernel(float* __restrict__ mask, float* __restrict__ noobj,
                                 float* __restrict__ tbox, unsigned* __restrict__ tcls) {
  int i = blockIdx.x * blockDim.x + threadIdx.x;
  if (i < NCELL) {
    mask[i]  = 0.0f;
    noobj[i] = 1.0f;
#pragma unroll
    for (int k = 0; k < 4; ++k) {
      tbox[(size_t)i * 4 + k] = 0.0f;
      tcls[(size_t)i * 4 + k] = 0u;
    }
  }
}

// ---------------- kernel B: encode targets (scatter) ----------------
__global__ void yolo_encode_kernel(const float* __restrict__ tgt,
                                   float* __restrict__ mask, float* __restrict__ noobj,
                                   float* __restrict__ tbox, unsigned* __restrict__ tcls) {
  int tid = blockIdx.x * blockDim.x + threadIdx.x;
  if (tid >= BATCH * TBOX) return;
  int b = tid / TBOX;
  const float* t = tgt + (size_t)tid * 5;
  float x = t[0], y = t[1], w = t[2], h = t[3], cf = t[4];
  if (!((x + y + w + h + cf) > 0.0f)) return;      // valid mask
  float gx = x * WG, gy = y * HG, gw = w * WG, gh = h * HG;
  int gi = min(max((int)gx, 0), WG - 1);
  int gj = min(max((int)gy, 0), HG - 1);
  const float aw[NA] = {0.57273f, 1.87446f, 3.33843f, 7.88282f, 9.77052f};
  const float ah[NA] = {0.677385f, 2.06253f, 5.47434f, 3.52778f, 9.16828f};
  int best = 0; float bestv = -1.0f;
#pragma unroll
  for (int a = 0; a < NA; ++a) {
    float inter = fminf(gw, aw[a]) * fminf(gh, ah[a]);
    float uni   = gw * gh + aw[a] * ah[a] - inter + 1e-16f;
    float iou   = inter / uni;
    if (iou > bestv) { bestv = iou; best = a; }     // first-max like argmax
    if (iou > 0.5f)                                  // IGNORE_THR
      noobj[((b * NA + a) * HG + gj) * WG + gi] = 0.0f;
  }
  int idx = ((b * NA + best) * HG + gj) * WG + gi;
  mask[idx] = 1.0f;                                  // tconf == mask
  tbox[(size_t)idx * 4 + 0] = gx - (float)gi;        // .set (last-writer, as JAX)
  tbox[(size_t)idx * 4 + 1] = gy - (float)gj;
  tbox[(size_t)idx * 4 + 2] = gw / aw[best];
  tbox[(size_t)idx * 4 + 3] = gh / ah[best];
  int cid = min(max((int)cf, 0), NCLS - 1);
  atomicOr(&tcls[(size_t)idx * 4 + (cid >> 5)], 1u << (cid & 31)); // one-hot .max
}

// ---------------- kernel C: sparse loss pass + WMMA reductions ----------------
__global__ void yolo_loss_kernel(const float* __restrict__ in,
                                 const float* __restrict__ mask,
                                 const float* __restrict__ noobj,
                                 const float* __restrict__ tbox,
                                 const unsigned* __restrict__ tcls,
                                 float* __restrict__ partials) {
  int idx = blockIdx.x * blockDim.x + threadIdx.x;
  float box_s = 0.0f, npos = 0.0f, obj_s = 0.0f, nob_s = 0.0f, cls_s = 0.0f, msum = 0.0f;
  if (idx < NCELL) {
    int b = idx / (NA * PLANE);
    int r = idx % (NA * PLANE);
    int a = r / PLANE;
    int cell = r % PLANE;
    const float* base = in + ((size_t)b * 425 + (size_t)a * 85) * PLANE + cell;
    float pc = sigm(base[4 * PLANE]);                // conf channel: always needed
    float m  = mask[idx];
    float nb = noobj[idx];
    obj_s = bce(pc * m, m);                          // LAMBDA_OBJ term (mean later)
    nob_s = bce(pc * nb, 0.0f);                      // LAMBDA_NOOBJ term
    msum  = m;
    if (m > 0.0f) {                                  // <=2.8% of cells: full 85ch work
      float px = sigm(base[0]);
      float py = sigm(base[PLANE]);
      float pw = expf(base[2 * PLANE]);
      float ph = expf(base[3 * PLANE]);
      const float* tb = tbox + (size_t)idx * 4;
      float ci = ciou_fn(px, py, pw, ph, tb[0], tb[1], tb[2], tb[3]);
      if (ci > 0.0f) { box_s = 1.0f - ci; npos = 1.0f; }
      unsigned w0 = tcls[(size_t)idx * 4 + 0];
      unsigned w1 = tcls[(size_t)idx * 4 + 1];
      unsigned w2 = tcls[(size_t)idx * 4 + 2];
      for (int c = 0; c < NCLS; ++c) {
        float pcl = sigm(base[(size_t)(5 + c) * PLANE]);
        unsigned bit = (c < 32) ? ((w0 >> c) & 1u)
                     : (c < 64) ? ((w1 >> (c - 32)) & 1u)
                                : ((w2 >> (c - 64)) & 1u);
        cls_s += bce(pcl, bit ? 1.0f : 0.0f);
      }
    }
  }
  // all lanes (full waves) reach here -> EXEC all ones for WMMA
  __shared__ float red[CBLK / 32][6];
  float acc[6] = {box_s, npos, obj_s, nob_s, cls_s, msum};
#pragma unroll
  for (int k = 0; k < 6; ++k) acc[k] = wave_sum32(acc[k]);
  int wave = threadIdx.x >> 5, lane = threadIdx.x & 31;
  if (lane == 0)
    for (int k = 0; k < 6; ++k) red[wave][k] = acc[k];
  __syncthreads();
  if (threadIdx.x == 0) {
    float s[6] = {0, 0, 0, 0, 0, 0};
    for (int w = 0; w < CBLK / 32; ++w)
      for (int k = 0; k < 6; ++k) s[k] += red[w][k];
    for (int k = 0; k < 6; ++k) partials[(size_t)blockIdx.x * 6 + k] = s[k];
  }
}

// ---------------- kernel D: finalize scalar ----------------
__global__ void yolo_final_kernel(const float* __restrict__ partials, int nblk,
                                  float* __restrict__ out) {
  __shared__ float red[CBLK / 32][6];
  float acc[6] = {0, 0, 0, 0, 0, 0};
  for (int p = threadIdx.x; p < nblk; p += blockDim.x)
    for (int k = 0; k < 6; ++k) acc[k] += partials[(size_t)p * 6 + k];
#pragma unroll
  for (int k = 0; k < 6; ++k) acc[k] = wave_sum32(acc[k]);
  int wave = threadIdx.x >> 5, lane = threadIdx.x & 31;
  if (lane == 0)
    for (int k = 0; k < 6; ++k) red[wave][k] = acc[k];
  __syncthreads();
  if (threadIdx.x == 0) {
    float s[6] = {0, 0, 0, 0, 0, 0};
    for (int w = 0; w < CBLK / 32; ++w)
      for (int k = 0; k < 6; ++k) s[k] += red[w][k];
    const float N = (float)NCELL;
    float box_loss   = 0.05f * s[0] / fmaxf(s[1], 1.0f);           // LAMBDA_COORD
    float obj_loss   = 10.0f * s[2] / N;                            // LAMBDA_OBJ * mean
    float noobj_loss = 1.0f  * s[3] / N;                            // LAMBDA_NOOBJ * mean
    float cls_loss   = 1.0f  * s[4] / fmaxf(s[5] * (float)NCLS, 1.0f);
    out[0] = (box_loss + obj_loss + noobj_loss + cls_loss) * (float)BATCH;
  }
}

// ---------------- launcher ----------------
extern "C" void kernel_launch(void* const* d_in, const int* in_sizes, int n_in,
                              void* d_out, int out_size, void* d_ws, size_t ws_size,
                              hipStream_t stream) {
  const float* input  = (const float*)d_in[0];   // (128,425,19,19) f32
  const float* target = (const float*)d_in[1];   // (128,50,5) f32
  float* out = (float*)d_out;

  // workspace layout (~9.3 MB)
  float*    mask     = (float*)d_ws;                          // NCELL
  float*    noobj    = mask + NCELL;                          // NCELL
  float*    tbox     = noobj + NCELL;                         // 4*NCELL
  unsigned* tcls     = (unsigned*)(tbox + (size_t)4 * NCELL); // 4*NCELL u32
  float*    partials = (float*)(tcls + (size_t)4 * NCELL);    // NBLK*6

  yolo_init_kernel<<<NBLK, CBLK, 0, stream>>>(mask, noobj, tbox, tcls);
  yolo_encode_kernel<<<(BATCH * TBOX + CBLK - 1) / CBLK, CBLK, 0, stream>>>(
      target, mask, noobj, tbox, tcls);
  yolo_loss_kernel<<<NBLK, CBLK, 0, stream>>>(input, mask, noobj, tbox, tcls, partials);
  yolo_final_kernel<<<1, CBLK, 0, stream>>>(partials, NBLK, out);
}